// Img_warp_loss_53936199303424
// MI455X (gfx1250) — compile-verified
//
#include <hip/hip_runtime.h>
#include <math.h>

// Problem constants (from reference): B=8, C=3, H=512, W=960
#define BDIM 8
#define CDIM 3
#define HDIM 512
#define WDIM 960
#define HWSZ (HDIM * WDIM)              // 491520
#define NPIX (BDIM * HWSZ)              // 3,932,160 pixels
#define NELEM (CDIM * NPIX)             // 11,796,480 output elements

#define NBLOCKS 1024
#define NTHREADS 256

typedef __attribute__((ext_vector_type(2))) float v2f;
typedef __attribute__((ext_vector_type(8))) float v8f;

// ---------------------------------------------------------------------------
// Kernel 1: warp + mask + store img_warp_mask + per-block partial |diff| sums.
// One thread handles one pixel (all 3 channels) so disp/left/out traffic is
// touched exactly once. Invalid bilinear taps are skipped (identical to the
// reference's clamp-then-multiply-by-zero). left_img is only read where the
// mask is 1 (elsewhere both operands of |a-b| are exactly 0).
// ---------------------------------------------------------------------------
__global__ __launch_bounds__(NTHREADS) void warp_loss_main(
    const float* __restrict__ disp,   // [B,2,H,W]
    const float* __restrict__ left,   // [B,C,H,W]
    const float* __restrict__ right,  // [B,C,H,W]
    float* __restrict__ out,          // [1 + B*C*H*W]; pixels start at +1
    float* __restrict__ partials)     // [NBLOCKS]
{
  float tsum = 0.0f;
  const int stride = gridDim.x * blockDim.x;
  for (int p = blockIdx.x * blockDim.x + threadIdx.x; p < NPIX; p += stride) {
    const int b  = p / HWSZ;
    const int hw = p - b * HWSZ;
    const int w  = hw % WDIM;

    const float d0 = disp[(size_t)(b * 2 + 0) * HWSZ + hw];
    const float d1 = disp[(size_t)(b * 2 + 1) * HWSZ + hw];
    const float gx = (float)w - d0;
    const float gy = (float)w - d1;
    const float ix = ((gx + 1.0f) * (float)WDIM - 1.0f) * 0.5f;
    const float iy = ((gy + 1.0f) * (float)HDIM - 1.0f) * 0.5f;

    const float x0 = floorf(ix), y0 = floorf(iy);
    const float x1 = x0 + 1.0f,  y1 = y0 + 1.0f;
    const float wx1 = ix - x0, wx0 = 1.0f - wx1;
    const float wy1 = iy - y0, wy0 = 1.0f - wy1;

    const float xs[2]  = {x0, x1};
    const float ys[2]  = {y0, y1};
    const float wxs[2] = {wx0, wx1};
    const float wys[2] = {wy0, wy1};

    float acc0 = 0.0f, acc1 = 0.0f, acc2 = 0.0f, msum = 0.0f;
    const size_t rbase = (size_t)b * (CDIM * HWSZ);
#pragma unroll
    for (int ty = 0; ty < 2; ++ty) {
#pragma unroll
      for (int tx = 0; tx < 2; ++tx) {
        const float xf = xs[tx], yf = ys[ty];
        const float wt = wxs[tx] * wys[ty];
        if (xf >= 0.0f && xf <= (float)(WDIM - 1) &&
            yf >= 0.0f && yf <= (float)(HDIM - 1)) {
          const int xi = (int)xf;
          const int yi = (int)yf;
          const size_t off = rbase + (size_t)yi * WDIM + xi;
          acc0 += right[off]            * wt;
          acc1 += right[off + HWSZ]     * wt;
          acc2 += right[off + 2 * HWSZ] * wt;
          msum += wt;
        }
      }
    }

    const float m = (msum < 0.999f) ? 0.0f : 1.0f;
    const float o0 = acc0 * m, o1 = acc1 * m, o2 = acc2 * m;
    const size_t obase = 1 + (size_t)(b * CDIM) * HWSZ + hw;
    out[obase]            = o0;
    out[obase + HWSZ]     = o1;
    out[obase + 2 * HWSZ] = o2;

    if (m != 0.0f) {
      const size_t lbase = (size_t)(b * CDIM) * HWSZ + hw;
      tsum += fabsf(o0 - left[lbase]);
      tsum += fabsf(o1 - left[lbase + HWSZ]);
      tsum += fabsf(o2 - left[lbase + 2 * HWSZ]);
    }
  }

  // Deterministic intra-block tree reduction in LDS.
  __shared__ float red[NTHREADS];
  red[threadIdx.x] = tsum;
  __syncthreads();
#pragma unroll
  for (int s = NTHREADS / 2; s > 0; s >>= 1) {
    if (threadIdx.x < s) red[threadIdx.x] += red[threadIdx.x + s];
    __syncthreads();
  }
  if (threadIdx.x == 0) partials[blockIdx.x] = red[0];
}

// ---------------------------------------------------------------------------
// Kernel 2: single wave (32 lanes, EXEC all ones). Each lane folds 32 of the
// 1024 partials into 2 registers (= the 64 elements of a 16x4 f32 A matrix),
// then V_WMMA_F32_16X16X4_F32 with an all-ones B computes all row sums; the
// per-lane sum of the 8 D VGPRs covers rows 0-7 (lanes 0-15) or 8-15
// (lanes 16-31), so one shfl_xor(16) yields the exact total. Fixed order
// every call => deterministic. No atomics.
// ---------------------------------------------------------------------------
__global__ void warp_loss_finish(const float* __restrict__ partials,
                                 float* __restrict__ out)
{
  const int lane = threadIdx.x;  // 0..31, one wave32
  float a0 = 0.0f, a1 = 0.0f;
#pragma unroll
  for (int j = 0; j < 16; ++j)  a0 += partials[j * 32 + lane];
#pragma unroll
  for (int j = 16; j < 32; ++j) a1 += partials[j * 32 + lane];

  v2f A;  A.x = a0;  A.y = a1;
  v2f Bv; Bv.x = 1.0f; Bv.y = 1.0f;   // all-ones B matrix (layout-independent)
  v8f Cv = {};
  // 8 args: (neg_a, A, neg_b, B, c_mod, C, reuse_a, reuse_b)
  v8f D = __builtin_amdgcn_wmma_f32_16x16x4_f32(
      /*neg_a=*/false, A, /*neg_b=*/false, Bv,
      /*c_mod=*/(short)0, Cv, /*reuse_a=*/false, /*reuse_b=*/false);

  float s = D[0] + D[1] + D[2] + D[3] + D[4] + D[5] + D[6] + D[7];
  s += __shfl_xor(s, 16, 32);          // combine rows 0-7 with rows 8-15
  if (lane == 0) out[0] = s / (float)NELEM;
}

extern "C" void kernel_launch(void* const* d_in, const int* in_sizes, int n_in,
                              void* d_out, int out_size, void* d_ws, size_t ws_size,
                              hipStream_t stream) {
  const float* disp  = (const float*)d_in[0];   // [B,2,H,W]
  const float* left  = (const float*)d_in[1];   // [B,C,H,W]
  const float* right = (const float*)d_in[2];   // [B,C,H,W]
  float* out = (float*)d_out;                   // [1 + B*C*H*W]
  float* partials = (float*)d_ws;               // NBLOCKS floats (4 KB)

  warp_loss_main<<<NBLOCKS, NTHREADS, 0, stream>>>(disp, left, right, out, partials);
  warp_loss_finish<<<1, 32, 0, stream>>>(partials, out);
}